// RPN_20066087207273
// MI455X (gfx1250) — compile-verified
//
#include <hip/hip_runtime.h>
#include <hip/hip_bf16.h>

// ---------------------------------------------------------------------------
// RPN head for MI455X (gfx1250): conv3x3 via V_WMMA_F32_16X16X4_F32 implicit
// GEMM (K split over 8 waves/block + LDS fragment reduction), then
// heads/decode, bitonic argsort, serial NMS, finalize.
// ---------------------------------------------------------------------------

typedef __attribute__((ext_vector_type(2))) float v2f;
typedef __attribute__((ext_vector_type(8))) float v8f;

#define IMG_WH 384
#define FEAT   24
#define POS    576           // FEAT*FEAT
#define CIN    512
#define MID    24
#define KANCH  9
#define NA     5184          // POS*KANCH
#define NPAD   8192          // sort padding (pow2 >= NA)

#define KSTEPS 1152          // 9 taps * (512/4)
#define NWAVE  8
#define KPW    (KSTEPS / NWAVE)   // 144 k-steps per wave

// workspace layout (4-byte units)
#define OFF_H      0                    // MID*POS = 13824 floats
#define OFF_PROBS  13824
#define OFF_SCORES (OFF_PROBS  + NA)
#define OFF_CX0    (OFF_SCORES + NA)
#define OFF_CY0    (OFF_CX0 + NA)
#define OFF_CX1    (OFF_CY0 + NA)
#define OFF_CY1    (OFF_CX1 + NA)
#define OFF_CWS    (OFF_CY1 + NA)
#define OFF_CHS    (OFF_CWS + NA)
#define OFF_VALID  (OFF_CHS + NA)       // int
#define OFF_ORDER  (OFF_VALID + NA)     // int
#define OFF_SX0    (OFF_ORDER + NA)
#define OFF_SY0    (OFF_SX0 + NA)
#define OFF_SX1    (OFF_SY0 + NA)
#define OFF_SY1    (OFF_SX1 + NA)
#define OFF_SAREA  (OFF_SY1 + NA)
#define OFF_SVALID (OFF_SAREA + NA)     // int
#define OFF_KEEPS  (OFF_SVALID + NA)    // int
#define OFF_KEEPO  (OFF_KEEPS + NA)     // int
#define OFF_WT     (OFF_KEEPO + NA)     // repacked weights [9][512][32] = 147456

// output layout (floats)
#define O_PROP  0                       // NA*4
#define O_CLS   (NA*4)                  // NA*2
#define O_FILT  (O_CLS + NA*2)          // NA*4
#define O_PROBK (O_FILT + NA*4)         // NA
#define O_LAB   (O_PROBK + NA)          // NA
#define O_KEEP  (O_LAB + NA)            // NA

// ---------------------------------------------------------------------------
// Kernel 0: repack W_rpn [24][512][3][3] -> Wt [tap][c][o(32, zero padded)].
// Makes the WMMA A-fragment loads lane-coalesced and removes padding branch.
// ---------------------------------------------------------------------------
__global__ __launch_bounds__(256) void repack_w_kernel(
    const float* __restrict__ Wr, float* __restrict__ Wt)
{
  const int idx = blockIdx.x * blockDim.x + threadIdx.x;   // [0, 9*512*32)
  if (idx >= 9 * CIN * 32) return;
  const int o   = idx & 31;
  const int c   = (idx >> 5) & (CIN - 1);
  const int tap = idx >> 14;                               // idx / (512*32)
  Wt[idx] = (o < MID) ? Wr[o * (CIN * 9) + c * 9 + tap] : 0.0f;
}

// ---------------------------------------------------------------------------
// Kernel 1: conv3x3 (512->24, pad 1) + bias + ReLU via fp32 WMMA.
// GEMM view: C[M=24 ch (2 tiles)][N=576 pos (36 tiles)] over K=512*9.
// 8 waves per block split the K loop; partial C fragments reduced in LDS.
// ---------------------------------------------------------------------------
__global__ __launch_bounds__(256) void conv_wmma_kernel(
    const float* __restrict__ X,       // [512][24][24]
    const float* __restrict__ Wt,      // [9][512][32] repacked
    const float* __restrict__ Br,      // [24]
    float* __restrict__ H)             // [24][576]
{
  __shared__ float red[NWAVE * 512];   // 8 waves x (2 tiles x 8 regs x 32 lanes)

  const int wave = threadIdx.x >> 5;
  const int lane = threadIdx.x & 31;
  const int half = lane >> 4;          // 0 -> K pair {k,k+1}, 1 -> {k+2,k+3}
  const int l    = lane & 15;
  const int p0   = blockIdx.x * 16;    // N-tile base (positions)

  const int n  = p0 + l;               // this lane's B column (position)
  const int py = n / FEAT;
  const int px = n % FEAT;

  v8f acc0 = {};                       // M tile 0: out ch 0..15
  v8f acc1 = {};                       // M tile 1: out ch 16..31 (24..31 zero)

  const int t0 = wave * KPW;
  for (int t = t0; t < t0 + KPW; ++t) {
    const int tap = t >> 7;            // 0..8
    const int kq  = (t & 127) << 2;    // K base, step 4
    const int dy  = tap / 3, dx = tap % 3;
    const int yy  = py + dy - 1;
    const int xx  = px + dx - 1;
    const bool inb = (yy >= 0) & (yy < FEAT) & (xx >= 0) & (xx < FEAT);

    const int ka = kq + half * 2;      // this half-wave's K pair
    const float* wt = Wt + (tap * CIN + ka) * 32;  // + c*32 + o
    v2f a0, a1, b;
    a0.x = wt[l];                      // W[o=l   ][ka]
    a0.y = wt[32 + l];                 // W[o=l   ][ka+1]
    a1.x = wt[16 + l];                 // W[o=l+16][ka]
    a1.y = wt[32 + 16 + l];            // W[o=l+16][ka+1]
    const float* xcol = X + yy * FEAT + xx;
    b.x = inb ? xcol[(ka    ) * POS] : 0.0f;
    b.y = inb ? xcol[(ka + 1) * POS] : 0.0f;

    acc0 = __builtin_amdgcn_wmma_f32_16x16x4_f32(
        false, a0, false, b, (short)0, acc0, false, false);
    acc1 = __builtin_amdgcn_wmma_f32_16x16x4_f32(
        false, a1, false, b, (short)0, acc1, false, false);
  }

  // park partial fragments in LDS: red[wave][tile*256 + r*32 + lane]
  float* myred = red + wave * 512;
#pragma unroll
  for (int r = 0; r < 8; ++r) {
    myred[r * 32 + lane]       = acc0[r];
    myred[256 + r * 32 + lane] = acc1[r];
  }
  __syncthreads();

  // reduce 8 partials; 512 outputs, 2 per thread; then bias + ReLU + store.
#pragma unroll
  for (int q = 0; q < 2; ++q) {
    const int e = q * 256 + threadIdx.x;     // tile*256 + r*32 + lane
    float s = 0.0f;
#pragma unroll
    for (int w = 0; w < NWAVE; ++w) s += red[w * 512 + e];

    const int tile = e >> 8;
    const int r    = (e >> 5) & 7;
    const int el   = e & 31;
    const int m    = r + (el >> 4) * 8;      // C layout: VGPR r -> M=r / r+8
    const int o    = tile * 16 + m;
    const int ncol = p0 + (el & 15);
    if (o < MID) {
      const float v = s + Br[o];
      H[o * POS + ncol] = v > 0.0f ? v : 0.0f;
    }
  }
}

// ---------------------------------------------------------------------------
// Kernel 2: heads (cls/reg), box decode, clip, softmax, validity, scores.
// ---------------------------------------------------------------------------
__global__ __launch_bounds__(256) void heads_decode_kernel(
    const float* __restrict__ ws,      // for H
    float* __restrict__ wso,           // mutable ws
    const float* __restrict__ Wcls,    // [18][24]
    const float* __restrict__ bcls,    // [18]
    const float* __restrict__ Wreg,    // [36][24]
    const float* __restrict__ breg,    // [36]
    const float* __restrict__ anchors, // [NA][4]
    const unsigned char* __restrict__ vmask, // [NA] bool
    float* __restrict__ out)
{
  const int nidx = blockIdx.x * blockDim.x + threadIdx.x;
  if (nidx >= NA) return;

  const float* H = ws + OFF_H;
  const int p = nidx / KANCH;          // spatial position
  const int k = nidx % KANCH;          // anchor kind

  float l0 = bcls[2 * k], l1 = bcls[2 * k + 1];
  float t0 = breg[4 * k], t1 = breg[4 * k + 1];
  float t2 = breg[4 * k + 2], t3 = breg[4 * k + 3];
#pragma unroll 4
  for (int c = 0; c < MID; ++c) {
    const float hv = H[c * POS + p];
    l0 += hv * Wcls[(2 * k    ) * MID + c];
    l1 += hv * Wcls[(2 * k + 1) * MID + c];
    t0 += hv * Wreg[(4 * k    ) * MID + c];
    t1 += hv * Wreg[(4 * k + 1) * MID + c];
    t2 += hv * Wreg[(4 * k + 2) * MID + c];
    t3 += hv * Wreg[(4 * k + 3) * MID + c];
  }

  const float ax = anchors[nidx * 4 + 0];
  const float ay = anchors[nidx * 4 + 1];
  const float aw = anchors[nidx * 4 + 2];
  const float ah = anchors[nidx * 4 + 3];

  const float pxc = ax + aw * t0;
  const float pyc = ay + ah * t1;
  const float pw  = aw * expf(t2);
  const float ph  = ah * expf(t3);

  out[O_PROP + nidx * 4 + 0] = pxc;
  out[O_PROP + nidx * 4 + 1] = pyc;
  out[O_PROP + nidx * 4 + 2] = pw;
  out[O_PROP + nidx * 4 + 3] = ph;
  out[O_CLS + nidx * 2 + 0] = l0;
  out[O_CLS + nidx * 2 + 1] = l1;

  float x0 = pxc - 0.5f * (pw - 1.0f);
  float y0 = pyc - 0.5f * (ph - 1.0f);
  float x1 = pw + x0 - 1.0f;
  float y1 = ph + y0 - 1.0f;
  const float lim = (float)(IMG_WH - 1);
  x0 = fminf(fmaxf(x0, 0.0f), lim);
  x1 = fminf(fmaxf(x1, 0.0f), lim);
  y0 = fminf(fmaxf(y0, 0.0f), lim);
  y1 = fminf(fmaxf(y1, 0.0f), lim);
  const float wsd = x1 - x0 + 1.0f;
  const float hsd = y1 - y0 + 1.0f;

  const float m = fmaxf(l0, l1);
  const float e0 = expf(l0 - m), e1 = expf(l1 - m);
  const float prob = e1 / (e0 + e1);

  const int valid = (vmask[nidx] != 0) && (wsd >= 16.0f) && (hsd >= 16.0f);

  wso[OFF_PROBS  + nidx] = prob;
  wso[OFF_SCORES + nidx] = valid ? prob : -1.0f;
  wso[OFF_CX0 + nidx] = x0;
  wso[OFF_CY0 + nidx] = y0;
  wso[OFF_CX1 + nidx] = x1;
  wso[OFF_CY1 + nidx] = y1;
  wso[OFF_CWS + nidx] = wsd;
  wso[OFF_CHS + nidx] = hsd;
  ((int*)wso)[OFF_VALID + nidx] = valid;
}

// ---------------------------------------------------------------------------
// Kernel 3: bitonic argsort (descending score) in LDS; gather sorted boxes.
// ---------------------------------------------------------------------------
__global__ __launch_bounds__(1024) void sort_kernel(float* __restrict__ ws)
{
  __shared__ unsigned long long keys[NPAD];   // 64 KB LDS (of 320 KB/WGP)
  const int tid = threadIdx.x;

  const float* SC = ws + OFF_SCORES;
#pragma unroll
  for (int e = 0; e < NPAD / 1024; ++e) {
    const int i = tid + e * 1024;
    unsigned long long key;
    if (i < NA) {
      unsigned u = __float_as_uint(SC[i]);
      u = (u & 0x80000000u) ? ~u : (u | 0x80000000u);  // ascending-orderable
      u = ~u;                                          // descending
      key = ((unsigned long long)u << 32) | (unsigned)i;
    } else {
      key = ~0ull;                                     // pad to the end
    }
    keys[i] = key;
  }
  __syncthreads();

  for (int k = 2; k <= NPAD; k <<= 1) {
    for (int j = k >> 1; j > 0; j >>= 1) {
#pragma unroll
      for (int e = 0; e < NPAD / 1024; ++e) {
        const int i = tid + e * 1024;
        const int ixj = i ^ j;
        if (ixj > i) {
          const bool up = ((i & k) == 0);
          const unsigned long long a = keys[i];
          const unsigned long long b = keys[ixj];
          if ((a > b) == up) { keys[i] = b; keys[ixj] = a; }
        }
      }
      __syncthreads();
    }
  }

  int* ORDER = (int*)ws + OFF_ORDER;
  for (int s = tid; s < NA; s += 1024) {
    const int orig = (int)(unsigned)keys[s];
    ORDER[s] = orig;
    ws[OFF_SX0 + s] = ws[OFF_CX0 + orig];
    ws[OFF_SY0 + s] = ws[OFF_CY0 + orig];
    ws[OFF_SX1 + s] = ws[OFF_CX1 + orig];
    ws[OFF_SY1 + s] = ws[OFF_CY1 + orig];
    ws[OFF_SAREA + s] = ws[OFF_CWS + orig] * ws[OFF_CHS + orig];
    ((int*)ws)[OFF_SVALID + s] = ((int*)ws)[OFF_VALID + orig];
  }
}

// ---------------------------------------------------------------------------
// Kernel 4: serial NMS scan (single block); suppression row parallelized.
// ---------------------------------------------------------------------------
__global__ __launch_bounds__(256) void nms_kernel(float* __restrict__ ws)
{
  __shared__ unsigned rem[(NA + 31) / 32];
  __shared__ int cur;
  for (int t = threadIdx.x; t < (NA + 31) / 32; t += 256) rem[t] = 0u;
  __syncthreads();

  const float* SX0 = ws + OFF_SX0;
  const float* SY0 = ws + OFF_SY0;
  const float* SX1 = ws + OFF_SX1;
  const float* SY1 = ws + OFF_SY1;
  const float* SAR = ws + OFF_SAREA;
  const int* SV = (int*)ws + OFF_SVALID;
  int* KEEPS = (int*)ws + OFF_KEEPS;

  for (int i = 0; i < NA; ++i) {
    if (threadIdx.x == 0) {
      const int kv = SV[i] && !((rem[i >> 5] >> (i & 31)) & 1u);
      cur = kv;
      KEEPS[i] = kv;
    }
    __syncthreads();
    if (cur) {
      const float x0i = SX0[i], y0i = SY0[i], x1i = SX1[i], y1i = SY1[i];
      const float ai = SAR[i];
      for (int j = i + 1 + (int)threadIdx.x; j < NA; j += 256) {
        float iw = fminf(x1i, SX1[j]) - fmaxf(x0i, SX0[j]) + 1.0f;
        float ih = fminf(y1i, SY1[j]) - fmaxf(y0i, SY0[j]) + 1.0f;
        iw = fmaxf(iw, 0.0f);
        ih = fmaxf(ih, 0.0f);
        const float inter = iw * ih;
        const float iou = inter / (ai + SAR[j] - inter);
        if (iou > 0.7f) atomicOr(&rem[j >> 5], 1u << (j & 31));
      }
    }
    __syncthreads();
  }
}

// ---------------------------------------------------------------------------
// Kernel 5: scatter keep back to original anchor order.
// ---------------------------------------------------------------------------
__global__ __launch_bounds__(256) void scatter_kernel(float* __restrict__ ws)
{
  const int s = blockIdx.x * blockDim.x + threadIdx.x;
  if (s >= NA) return;
  const int* ORDER = (int*)ws + OFF_ORDER;
  const int* KEEPS = (int*)ws + OFF_KEEPS;
  int* KEEPO = (int*)ws + OFF_KEEPO;
  KEEPO[ORDER[s]] = KEEPS[s];
}

// ---------------------------------------------------------------------------
// Kernel 6: finalize filtered boxes / kept probs / kept labels / keep flags.
// ---------------------------------------------------------------------------
__global__ __launch_bounds__(256) void finalize_kernel(
    const float* __restrict__ ws,
    const int* __restrict__ labels,
    float* __restrict__ out)
{
  const int nidx = blockIdx.x * blockDim.x + threadIdx.x;
  if (nidx >= NA) return;
  const int keep = ((const int*)ws)[OFF_KEEPO + nidx];
  const float kf = keep ? 1.0f : 0.0f;
  const float x0 = ws[OFF_CX0 + nidx];
  const float y0 = ws[OFF_CY0 + nidx];
  const float wsd = ws[OFF_CWS + nidx];
  const float hsd = ws[OFF_CHS + nidx];
  out[O_FILT + nidx * 4 + 0] = (x0 + 0.5f * (wsd - 1.0f)) * kf;
  out[O_FILT + nidx * 4 + 1] = (y0 + 0.5f * (hsd - 1.0f)) * kf;
  out[O_FILT + nidx * 4 + 2] = wsd * kf;
  out[O_FILT + nidx * 4 + 3] = hsd * kf;
  out[O_PROBK + nidx] = ws[OFF_PROBS + nidx] * kf;
  out[O_LAB + nidx] = keep ? (float)labels[nidx] : 0.0f;
  out[O_KEEP + nidx] = kf;
}

// ---------------------------------------------------------------------------
extern "C" void kernel_launch(void* const* d_in, const int* in_sizes, int n_in,
                              void* d_out, int out_size, void* d_ws, size_t ws_size,
                              hipStream_t stream) {
  const float* X       = (const float*)d_in[0];            // (1,512,24,24)
  const int*   labels  = (const int*)d_in[1];              // (NA,)
  const float* Wrpn    = (const float*)d_in[2];            // (24,512,3,3)
  const float* brpn    = (const float*)d_in[3];            // (24,)
  const float* Wcls    = (const float*)d_in[4];            // (18,24)
  const float* bcls    = (const float*)d_in[5];            // (18,)
  const float* Wreg    = (const float*)d_in[6];            // (36,24)
  const float* breg    = (const float*)d_in[7];            // (36,)
  const float* anchors = (const float*)d_in[8];            // (NA,4)
  const unsigned char* vmask = (const unsigned char*)d_in[9]; // (NA,) bool

  float* ws  = (float*)d_ws;
  float* out = (float*)d_out;

  // 0) repack conv weights into WMMA-friendly, zero-padded layout
  repack_w_kernel<<<(9 * CIN * 32 + 255) / 256, 256, 0, stream>>>(
      Wrpn, ws + OFF_WT);

  // 1) conv3x3 + ReLU via fp32 WMMA: 36 N-tiles x 8 K-split waves
  conv_wmma_kernel<<<POS / 16, 256, 0, stream>>>(X, ws + OFF_WT, brpn,
                                                 ws + OFF_H);

  // 2) heads + decode + scores
  const int blk = (NA + 255) / 256;
  heads_decode_kernel<<<blk, 256, 0, stream>>>(ws, ws, Wcls, bcls, Wreg, breg,
                                               anchors, vmask, out);

  // 3) argsort (descending) + gather sorted boxes
  sort_kernel<<<1, 1024, 0, stream>>>(ws);

  // 4) serial NMS
  nms_kernel<<<1, 256, 0, stream>>>(ws);

  // 5) scatter keep to original order
  scatter_kernel<<<blk, 256, 0, stream>>>(ws);

  // 6) finalize outputs
  finalize_kernel<<<blk, 256, 0, stream>>>(ws, labels, out);
}